// ImplicitNet_79602923864103
// MI455X (gfx1250) — compile-verified
//
#include <hip/hip_runtime.h>

typedef __attribute__((ext_vector_type(2))) float v2f;
typedef __attribute__((ext_vector_type(8))) float v8f;
typedef __attribute__((ext_vector_type(4))) unsigned int u32x4;
typedef __attribute__((ext_vector_type(8))) int i32x8;
typedef __attribute__((ext_vector_type(4))) int i32x4;

#define ACT_S 260            // activation row stride (dwords), 260%64=4 -> conflict-free banks
#define KB    32             // K-chunk per TDM load / compute phase
#define WCH_S 36             // weight-chunk row stride (dwords) = KB + 4 pad (TDM pad_enable)
#define PAD_IV 4             // pad interval code: 2^(4+1) = 32 dwords
#define PAD_AMT 3            // pad amount code: 3+1 = 4 dwords
#define BLOCK 128            // 4 wave32s
#define PTS_WG 64            // 16 points per wave
#define PBATCH 50000
#define OUT_LAST 257
#define INV_SQRT2 0.70710678118654752f

#define WBUF_FLOATS (272 * WCH_S)               // one weight buffer (max 17 n-tiles)
#define WCH_BASE_FLOATS (PTS_WG * ACT_S)        // weight buffers start after act
#define SMEM_FLOATS (WCH_BASE_FLOATS + 2 * WBUF_FLOATS)

__device__ __forceinline__ float softplus100(float y) {
  // log(1+exp(100y))/100, numerically stable
  float t = 100.0f * y;
  return 0.01f * (fmaxf(t, 0.0f) + log1pf(__expf(-fabsf(t))));
}

// Issue one TDM 2D tile load: W[0:rows, kb:kb+KB] (row stride = row_stride elements,
// fp32) -> LDS at lds_off_bytes with 4-dword row padding.  OOB reads (cols beyond
// rem_k, rows beyond rows_real) return zero => all layer-dim padding is free.
__device__ __forceinline__ void tdm_load_w(const float* gsrc, unsigned lds_off_bytes,
                                           int rem_k, int row_stride,
                                           int rows_real, int rows_pad) {
  unsigned long long ga = (unsigned long long)(uintptr_t)gsrc;
  u32x4 g0;
  g0[0] = 1u;                                            // count=1, user descriptor
  g0[1] = lds_off_bytes;                                 // lds_addr (bytes)
  g0[2] = (unsigned)(ga & 0xFFFFFFFFu);                  // global_addr[31:0]
  g0[3] = (unsigned)((ga >> 32) & 0x01FFFFFFu) | (2u << 30);  // addr[56:32] | type=2
  i32x8 g1;
  g1[0] = (2 << 16)            // data_size = 4 bytes
        | (1 << 20)            // pad_enable
        | (PAD_IV << 22)       // pad interval: 32 dwords
        | (PAD_AMT << 25);     // pad amount: 4 dwords
  g1[1] = (rem_k & 0xFFFF) << 16;                               // tensor_dim0[15:0]
  g1[2] = ((unsigned)rem_k >> 16) | ((rows_real & 0xFFFF) << 16); // dim0 hi | dim1 lo
  g1[3] = (KB << 16);                                           // dim1 hi(=0) | tile_dim0
  g1[4] = rows_pad & 0xFFFF;                                    // tile_dim1 (tile_dim2=0)
  g1[5] = row_stride;                                           // tensor_dim0_stride lo32
  g1[6] = 0;                                                    // stride hi | dim1_stride lo
  g1[7] = 0;
  i32x4 gz4 = {0, 0, 0, 0};
  i32x8 gz8 = {0, 0, 0, 0, 0, 0, 0, 0};
  // 6-arg toolchain variant: (g0, g1, g2, g3, g4, cpol)
  __builtin_amdgcn_tensor_load_to_lds(g0, g1, gz4, gz4, gz8, 0);
}

// One linear layer:  act(64 x KPAD in LDS) @ W^T (+bias) -> act or global out.
// Weights double-buffered in LDS via TDM; compute overlaps next-chunk DMA.
template<int NT, int KREAL, int KPAD, bool DO_SP, bool SKIP_FIX, bool TO_GLOBAL>
__device__ __forceinline__ void mlp_layer(
    float* __restrict__ act, float* __restrict__ wchbase,
    const float* __restrict__ Wl, const float* __restrict__ bl, int out_real,
    float* __restrict__ outg, long long row0, const float* __restrict__ input,
    int lane, int wave)
{
  constexpr int NPH  = KPAD / KB;        // phases (even: 2 or 8)
  constexpr int OUTP = NT * 16;
  const int m_base = wave * 16;
  const int nloc   = lane & 15;          // M (for A) / N (for B,C,D) within tile
  const int hi     = (lane >> 4) << 1;   // 0 (lanes 0-15) or 2 (lanes 16-31)

  v8f C[NT];
  #pragma unroll
  for (int nt = 0; nt < NT; ++nt) {
    const int o = nt * 16 + nloc;
    const float bv = (o < out_real) ? bl[o] : 0.0f;   // bias broadcast along M
    #pragma unroll
    for (int r = 0; r < 8; ++r) C[nt][r] = bv;
  }

  // Prologue: chunk 0 -> buffer 0 (previous layer ended on buffer 1; all waves
  // passed that phase's barrier before wave 0 could reach here).
  if (wave == 0)
    tdm_load_w(Wl, (unsigned)(WCH_BASE_FLOATS * 4), KREAL, KREAL, out_real, OUTP);

  for (int ph = 0; ph < NPH; ++ph) {
    const int kb = ph * KB;
    __syncthreads();   // phase ph-1 compute done everywhere -> buffer (ph+1)&1 free
    if (wave == 0) {
      if (ph + 1 < NPH) {
        tdm_load_w(Wl + (kb + KB),
                   (unsigned)((WCH_BASE_FLOATS + ((ph + 1) & 1) * WBUF_FLOATS) * 4),
                   KREAL - (kb + KB), KREAL, out_real, OUTP);
        __builtin_amdgcn_s_wait_tensorcnt(1);   // current chunk (in-order) has landed
      } else {
        __builtin_amdgcn_s_wait_tensorcnt(0);   // last phase: all TDM done
      }
    }
    __syncthreads();   // chunk ph data visible to all waves

    const float* wch = wchbase + ((ph & 1) ? WBUF_FLOATS : 0);
    #pragma unroll
    for (int k2 = 0; k2 < KB; k2 += 4) {
      // A fragment (16x4 f32): lane holds X[m, k+hi], X[m, k+hi+1]
      const v2f a = *(const v2f*)&act[(m_base + nloc) * ACT_S + kb + k2 + hi];
      #pragma unroll
      for (int nt = 0; nt < NT; ++nt) {
        // B fragment (4x16 f32): lane holds W^T[k+hi, n], W^T[k+hi+1, n]
        const v2f b = *(const v2f*)&wch[(nt * 16 + nloc) * WCH_S + k2 + hi];
        C[nt] = __builtin_amdgcn_wmma_f32_16x16x4_f32(
            false, a, false, b, (short)0, C[nt], false, false);
      }
    }
  }

  // D layout: vgpr r, lane L -> (M = r + 8*(L>=16), N = L&15)
  const int mrow = m_base + ((lane >> 4) << 3);
  #pragma unroll
  for (int nt = 0; nt < NT; ++nt) {
    const int col = nt * 16 + nloc;
    #pragma unroll
    for (int r = 0; r < 8; ++r) {
      float v = C[nt][r];
      if (DO_SP)    v = softplus100(v);
      if (SKIP_FIX) v *= INV_SQRT2;
      const int mm = mrow + r;
      if (TO_GLOBAL) {
        if (col < out_real) outg[(row0 + mm) * OUT_LAST + col] = v;
      } else {
        act[mm * ACT_S + col] = v;        // wave-private rows: no barrier needed
      }
    }
  }
  if (SKIP_FIX) {
    // Skip connection: overwrite padded cols 253..255 with raw xyz * 1/sqrt(2)
    for (int idx = lane; idx < 48; idx += 32) {
      const int lp = idx / 3, c3 = idx % 3;
      const long long n = row0 + m_base + lp;
      act[(m_base + lp) * ACT_S + 253 + c3] = input[n * 3 + c3] * INV_SQRT2;
    }
  }
}

__global__ __launch_bounds__(BLOCK) void implicit_net_wmma_f32(
    const float* __restrict__ input, const float* __restrict__ cond,
    const float* __restrict__ W0, const float* __restrict__ b0,
    const float* __restrict__ W1, const float* __restrict__ b1,
    const float* __restrict__ W2, const float* __restrict__ b2,
    const float* __restrict__ W3, const float* __restrict__ b3,
    const float* __restrict__ W4, const float* __restrict__ b4,
    const float* __restrict__ W5, const float* __restrict__ b5,
    const float* __restrict__ W6, const float* __restrict__ b6,
    const float* __restrict__ W7, const float* __restrict__ b7,
    const float* __restrict__ W8, const float* __restrict__ b8,
    float* __restrict__ out)
{
  extern __shared__ float smem[];
  float* act = smem;                         // 64 x 260 floats
  float* wch = smem + WCH_BASE_FLOATS;       // 2 x (272 x 36) floats, TDM-filled

  const int tid  = (int)threadIdx.x;
  const int lane = tid & 31;
  const int wave = tid >> 5;
  const int m_base = wave * 16;
  const long long row0 = (long long)blockIdx.x * PTS_WG;

  // Build layer-0 input rows: [xyz(3) | cond_pose(48) | zero-pad(13)]
  for (int idx = lane; idx < 16 * 64; idx += 32) {
    const int lp = idx >> 6;
    const int c  = idx & 63;
    const long long n = row0 + m_base + lp;
    const int b = (int)(n / PBATCH);
    float v = 0.0f;
    if (c < 3)       v = input[n * 3 + c];
    else if (c < 51) v = cond[b * 48 + (c - 3)];
    act[(m_base + lp) * ACT_S + c] = v;
  }

  mlp_layer<16,  51,  64, true,  false, false>(act, wch, W0, b0, 256, out, row0, input, lane, wave);
  mlp_layer<16, 256, 256, true,  false, false>(act, wch, W1, b1, 256, out, row0, input, lane, wave);
  mlp_layer<16, 256, 256, true,  false, false>(act, wch, W2, b2, 256, out, row0, input, lane, wave);
  mlp_layer<16, 256, 256, true,  true,  false>(act, wch, W3, b3, 253, out, row0, input, lane, wave);
  mlp_layer<16, 256, 256, true,  false, false>(act, wch, W4, b4, 256, out, row0, input, lane, wave);
  mlp_layer<16, 256, 256, true,  false, false>(act, wch, W5, b5, 256, out, row0, input, lane, wave);
  mlp_layer<16, 256, 256, true,  false, false>(act, wch, W6, b6, 256, out, row0, input, lane, wave);
  mlp_layer<16, 256, 256, true,  false, false>(act, wch, W7, b7, 256, out, row0, input, lane, wave);
  mlp_layer<17, 256, 256, false, false, true >(act, wch, W8, b8, 257, out, row0, input, lane, wave);
}

extern "C" void kernel_launch(void* const* d_in, const int* in_sizes, int n_in,
                              void* d_out, int out_size, void* d_ws, size_t ws_size,
                              hipStream_t stream) {
  (void)in_sizes; (void)n_in; (void)out_size; (void)d_ws; (void)ws_size;
  const float* input = (const float*)d_in[0];
  const float* cond  = (const float*)d_in[1];
  const float* W[9];
  const float* B[9];
  for (int l = 0; l < 9; ++l) {
    W[l] = (const float*)d_in[2 + 2 * l];
    B[l] = (const float*)d_in[3 + 2 * l];
  }
  float* out = (float*)d_out;

  const size_t smem = (size_t)SMEM_FLOATS * sizeof(float);  // ~141.5 KB -> 2 WGs/WGP
  (void)hipFuncSetAttribute((const void*)implicit_net_wmma_f32,
                            hipFuncAttributeMaxDynamicSharedMemorySize, (int)smem);

  const int num_wgs = 200000 / PTS_WG;   // 3125, exact
  implicit_net_wmma_f32<<<num_wgs, BLOCK, smem, stream>>>(
      input, cond,
      W[0], B[0], W[1], B[1], W[2], B[2], W[3], B[3], W[4], B[4],
      W[5], B[5], W[6], B[6], W[7], B[7], W[8], B[8],
      out);
}